// UncertaintyGAT_83064667504939
// MI455X (gfx1250) — compile-verified
//
#include <hip/hip_runtime.h>
#include <hip/hip_bf16.h>
#include <math.h>

typedef float v2f __attribute__((ext_vector_type(2)));
typedef float v8f __attribute__((ext_vector_type(8)));

#define SLOPE 0.2f

static inline int cdiv(long a, long b) { return (int)((a + b - 1) / b); }

// ---------------- concat(emb[node_ids], feat) -> [N,128], float4 ----------------
__global__ __launch_bounds__(256) void k_concat(const float* __restrict__ emb,
                                                const int* __restrict__ ids,
                                                const float* __restrict__ feat,
                                                float* __restrict__ out, int n) {
  int t = blockIdx.x * blockDim.x + threadIdx.x;       // one float4 per thread
  if (t >= n * 32) return;                             // 128/4 = 32 vec4 per node
  int node = t >> 5, q = t & 31;
  int f = q << 2;
  const float* sp = (f < 64) ? (emb + (size_t)ids[node] * 64 + f)
                             : (feat + (size_t)node * 64 + (f - 64));
  *(float4*)(out + (size_t)node * 128 + f) = *(const float4*)sp;
}

// ---------------- fp32 WMMA GEMM: C[M,Nn] = A[M,K] @ B[K,Nn] (+bias, relu) -----
// One wave per 16x16 output tile, K-loop of V_WMMA_F32_16X16X4_F32.
__global__ __launch_bounds__(256) void k_gemm_wmma_f32(const float* __restrict__ A,
                                                       const float* __restrict__ B,
                                                       const float* __restrict__ bias,
                                                       float* __restrict__ C,
                                                       int M, int K, int Nn, int relu) {
  int wave = threadIdx.x >> 5;
  int lane = threadIdx.x & 31;
  int ntc = Nn >> 4;
  int tile = blockIdx.x * 8 + wave;
  int total = (M >> 4) * ntc;
  if (tile >= total) return;             // wave-uniform: EXEC stays all-ones
  int tr = tile / ntc, tc = tile % ntc;
  int r16 = lane & 15;
  int hi = lane >> 4;                    // 0 or 1
  int kh = hi << 1;                      // 0 or 2
  const float* Arow = A + (size_t)(tr * 16 + r16) * K;
  const float* Bcol = B + tc * 16 + r16;
  v8f acc = {};
  for (int k = 0; k < K; k += 4) {
    v2f a, b;
    a = *(const v2f*)(Arow + k + kh);    // contiguous -> global_load_b64
    const float* Bp = Bcol + (size_t)(k + kh) * Nn;
    b.x = Bp[0];
    b.y = Bp[Nn];
    acc = __builtin_amdgcn_wmma_f32_16x16x4_f32(false, a, false, b, (short)0, acc,
                                                false, false);
  }
  int col = tc * 16 + r16;
  int rbase = tr * 16 + (hi << 3);
  float bv = bias ? bias[col] : 0.f;
#pragma unroll
  for (int i = 0; i < 8; i++) {
    float v = acc[i] + bv;
    if (relu) v = fmaxf(v, 0.f);
    C[(size_t)(rbase + i) * Nn + col] = v;
  }
}

// ---------------- per-node attention dots: el/er [N,H], float4 loads --------
__global__ __launch_bounds__(256) void k_eler(const float* __restrict__ z,
                                              const float* __restrict__ al,
                                              const float* __restrict__ ar,
                                              float* __restrict__ el, float* __restrict__ er,
                                              int n, int H, int D) {
  int t = blockIdx.x * blockDim.x + threadIdx.x;
  if (t >= n * H) return;
  int node = t / H, h = t % H;
  const float* zp = z + (size_t)node * H * D + (size_t)h * D;
  const float* alp = al + h * D;
  const float* arp = ar + h * D;
  float sl = 0.f, sr = 0.f;
  for (int d = 0; d < D; d += 4) {
    float4 zv = *(const float4*)(zp + d);
    float4 av = *(const float4*)(alp + d);
    float4 rv = *(const float4*)(arp + d);
    sl += zv.x * av.x + zv.y * av.y + zv.z * av.z + zv.w * av.w;
    sr += zv.x * rv.x + zv.y * rv.y + zv.z * rv.z + zv.w * rv.w;
  }
  el[t] = sl;
  er[t] = sr;
}

// ---------------- init: segment max/sum state + zero accumulator --------
__global__ __launch_bounds__(256) void k_init(float* __restrict__ mm, float* __restrict__ ss,
                                              float* __restrict__ acc, int nH, long nF) {
  long t = (long)blockIdx.x * blockDim.x + threadIdx.x;
  if (t < nH) { mm[t] = -INFINITY; ss[t] = 0.f; }
  long f4 = t << 2;                      // zero 4 floats per thread
  if (f4 + 3 < nF) {
    *(float4*)(acc + f4) = make_float4(0.f, 0.f, 0.f, 0.f);
  } else {
    for (long i = f4; i < nF; i++) acc[i] = 0.f;
  }
}

// ordered-bits float atomic max (works with -inf init)
__device__ __forceinline__ void atomicMaxF(float* addr, float value) {
  if (value >= 0.f)
    atomicMax((int*)addr, __float_as_int(value));
  else
    atomicMin((unsigned int*)addr, __float_as_uint(value));
}

// ---------------- edge pass 1: e = leaky_relu(el[src]+er[dst]); seg-max ----
// one thread per edge, both heads handled (index loads once)
__global__ __launch_bounds__(256) void k_edge_max(const int* __restrict__ src,
                                                  const int* __restrict__ dst,
                                                  const float* __restrict__ el,
                                                  const float* __restrict__ er,
                                                  float* __restrict__ ex, float* __restrict__ mm,
                                                  int E, int H) {
  int e = blockIdx.x * blockDim.x + threadIdx.x;
  if (e >= E) return;
  int s = src[e], d = dst[e];
  for (int h = 0; h < H; h++) {
    float v = el[s * H + h] + er[d * H + h];
    v = (v > 0.f) ? v : v * SLOPE;
    ex[(size_t)e * H + h] = v;
    atomicMaxF(&mm[d * H + h], v);
  }
}

// ---------------- edge pass 2: ex = exp(e - m[dst]); seg-sum ----------------
__global__ __launch_bounds__(256) void k_edge_expsum(const int* __restrict__ dst,
                                                     float* __restrict__ ex,
                                                     const float* __restrict__ mm,
                                                     float* __restrict__ ss, int E, int H) {
  int e = blockIdx.x * blockDim.x + threadIdx.x;
  if (e >= E) return;
  int d = dst[e];
  for (int h = 0; h < H; h++) {
    float v = __expf(ex[(size_t)e * H + h] - mm[d * H + h]);
    ex[(size_t)e * H + h] = v;
    atomicAdd(&ss[d * H + h], v);
  }
}

// ---------------- edge pass 3 (F=128): one WAVE per edge ------------------
// lane owns 4 contiguous features: float4 gather of z[src], 4 atomic adds.
// src/dst/ex/ss loaded once per wave instead of once per feature.
__global__ __launch_bounds__(256) void k_edge_agg_wave(const int* __restrict__ src,
                                                       const int* __restrict__ dst,
                                                       const float* __restrict__ ex,
                                                       const float* __restrict__ ss,
                                                       const float* __restrict__ z,
                                                       float* __restrict__ acc,
                                                       int E, int H, int D) {
  int e = blockIdx.x * 8 + (threadIdx.x >> 5);     // wave index == edge
  if (e >= E) return;                              // wave-uniform
  int lane = threadIdx.x & 31;
  int s = src[e], d = dst[e];
  int F = H * D;                                   // 128
  int f = lane << 2;                               // 4 floats per lane
  int h = f / D;                                   // head for this lane
  float alpha = ex[(size_t)e * H + h] / ss[(size_t)d * H + h];
  float4 zv = *(const float4*)(z + (size_t)s * F + f);
  float* ap = acc + (size_t)d * F + f;
  atomicAdd(ap + 0, alpha * zv.x);
  atomicAdd(ap + 1, alpha * zv.y);
  atomicAdd(ap + 2, alpha * zv.z);
  atomicAdd(ap + 3, alpha * zv.w);
}

// ---------------- edge pass 3 (F=1): scalar, one thread per edge ----------
__global__ __launch_bounds__(256) void k_edge_agg_scalar(const int* __restrict__ src,
                                                         const int* __restrict__ dst,
                                                         const float* __restrict__ ex,
                                                         const float* __restrict__ ss,
                                                         const float* __restrict__ z,
                                                         float* __restrict__ acc, int E) {
  int e = blockIdx.x * blockDim.x + threadIdx.x;
  if (e >= E) return;
  int s = src[e], d = dst[e];
  float alpha = ex[e] / ss[d];
  atomicAdd(&acc[d], alpha * z[s]);
}

// ---------------- bias (+relu) epilogue, float4 (F multiple of 4) ---------
__global__ __launch_bounds__(256) void k_bias_act4(const float* __restrict__ in,
                                                   const float* __restrict__ bias,
                                                   float* __restrict__ out,
                                                   long total4, int F, int relu) {
  long t = (long)blockIdx.x * blockDim.x + threadIdx.x;
  if (t >= total4) return;
  long f4 = t << 2;
  int f = (int)(f4 % F);
  float4 v = *(const float4*)(in + f4);
  float4 b = *(const float4*)(bias + f);
  v.x += b.x; v.y += b.y; v.z += b.z; v.w += b.w;
  if (relu) {
    v.x = fmaxf(v.x, 0.f); v.y = fmaxf(v.y, 0.f);
    v.z = fmaxf(v.z, 0.f); v.w = fmaxf(v.w, 0.f);
  }
  *(float4*)(out + f4) = v;
}

// ---------------- scalar bias (final layer, F=1) ----------------
__global__ __launch_bounds__(256) void k_bias_scalar(const float* __restrict__ in,
                                                     const float* __restrict__ bias,
                                                     float* __restrict__ out, int n) {
  int t = blockIdx.x * blockDim.x + threadIdx.x;
  if (t >= n) return;
  out[t] = in[t] + bias[0];
}

// ---------------- output layer GEMV: z=h@w (128->1), el=z*al, er=z*ar -------
__global__ __launch_bounds__(256) void k_out_proj(const float* __restrict__ h,
                                                  const float* __restrict__ w,
                                                  const float* __restrict__ alp,
                                                  const float* __restrict__ arp,
                                                  float* __restrict__ z, float* __restrict__ el,
                                                  float* __restrict__ er, int n) {
  int t = blockIdx.x * blockDim.x + threadIdx.x;
  if (t >= n) return;
  const float* hp = h + (size_t)t * 128;
  float acc = 0.f;
#pragma unroll 8
  for (int k = 0; k < 128; k += 4) {
    float4 hv = *(const float4*)(hp + k);
    float4 wv = *(const float4*)(w + k);
    acc += hv.x * wv.x + hv.y * wv.y + hv.z * wv.z + hv.w * wv.w;
  }
  z[t] = acc;
  el[t] = acc * alp[0];
  er[t] = acc * arp[0];
}

extern "C" void kernel_launch(void* const* d_in, const int* in_sizes, int n_in,
                              void* d_out, int out_size, void* d_ws, size_t ws_size,
                              hipStream_t stream) {
  const float* feat    = (const float*)d_in[0];
  const int*   nodeids = (const int*)d_in[1];
  const int*   src     = (const int*)d_in[2];
  const int*   dst     = (const int*)d_in[3];
  const float* emb     = (const float*)d_in[4];
  const float* dense_w = (const float*)d_in[5];
  const float* dense_b = (const float*)d_in[6];
  const float* g1_w  = (const float*)d_in[7];
  const float* g1_al = (const float*)d_in[8];
  const float* g1_ar = (const float*)d_in[9];
  const float* g1_b  = (const float*)d_in[10];
  const float* g2_w  = (const float*)d_in[11];
  const float* g2_al = (const float*)d_in[12];
  const float* g2_ar = (const float*)d_in[13];
  const float* g2_b  = (const float*)d_in[14];
  const float* out_w  = (const float*)d_in[15];
  const float* out_al = (const float*)d_in[16];
  const float* out_ar = (const float*)d_in[17];
  const float* out_b  = (const float*)d_in[18];
  float* outp = (float*)d_out;

  const int N = in_sizes[1];
  const int E = in_sizes[2];
  const int F = 128;   // H*D for GAT layers
  const int H = 2, D = 64;

  float* ws = (float*)d_ws;
  size_t o = 0;
  float* bufA = ws + o; o += (size_t)N * F;   // node features (layer input)
  float* bufB = ws + o; o += (size_t)N * F;   // z = h @ W
  float* bufC = ws + o; o += (size_t)N * F;   // aggregation accumulator
  float* el   = ws + o; o += (size_t)N * H;
  float* er   = ws + o; o += (size_t)N * H;
  float* mm   = ws + o; o += (size_t)N * H;
  float* ss   = ws + o; o += (size_t)N * H;
  float* ex   = ws + o; o += (size_t)E * H;
  (void)ws_size; (void)n_in; (void)out_size;

  const int B256 = 256;
  const int gemmBlocks = cdiv((long)(N / 16) * (F / 16), 8);

  // ---- stage 0: concat -> bufC; dense+relu (WMMA) -> bufA ----
  k_concat<<<cdiv((long)N * 32, B256), B256, 0, stream>>>(emb, nodeids, feat, bufC, N);
  k_gemm_wmma_f32<<<gemmBlocks, B256, 0, stream>>>(bufC, dense_w, dense_b, bufA, N, 128, F, 1);

  // ---- two GAT layers (H=2, D=64) ----
  const float* Ws[2]  = {g1_w, g2_w};
  const float* ALs[2] = {g1_al, g2_al};
  const float* ARs[2] = {g1_ar, g2_ar};
  const float* Bs[2]  = {g1_b, g2_b};
  for (int layer = 0; layer < 2; layer++) {
    // z = h @ W   (WMMA, no bias/relu)
    k_gemm_wmma_f32<<<gemmBlocks, B256, 0, stream>>>(bufA, Ws[layer], nullptr, bufB, N, F, F, 0);
    // attention logits per node
    k_eler<<<cdiv((long)N * H, B256), B256, 0, stream>>>(bufB, ALs[layer], ARs[layer], el, er, N, H, D);
    // init segment state + zero accumulator (4 floats per thread)
    k_init<<<cdiv((long)N * F / 4, B256), B256, 0, stream>>>(mm, ss, bufC, N * H, (long)N * F);
    // edge softmax (3 passes) + aggregation
    k_edge_max<<<cdiv(E, B256), B256, 0, stream>>>(src, dst, el, er, ex, mm, E, H);
    k_edge_expsum<<<cdiv(E, B256), B256, 0, stream>>>(dst, ex, mm, ss, E, H);
    k_edge_agg_wave<<<cdiv(E, 8), B256, 0, stream>>>(src, dst, ex, ss, bufB, bufC, E, H, D);
    // h_next = relu(acc + b)
    k_bias_act4<<<cdiv((long)N * F / 4, B256), B256, 0, stream>>>(bufC, Bs[layer], bufA,
                                                                  (long)N * F / 4, F, 1);
  }

  // ---- output GAT layer (H=1, D=1) ----
  k_out_proj<<<cdiv(N, B256), B256, 0, stream>>>(bufA, out_w, out_al, out_ar, bufB, el, er, N);
  k_init<<<cdiv((long)N / 4 + 1, B256), B256, 0, stream>>>(mm, ss, outp, N, (long)N);
  k_edge_max<<<cdiv(E, B256), B256, 0, stream>>>(src, dst, el, er, ex, mm, E, 1);
  k_edge_expsum<<<cdiv(E, B256), B256, 0, stream>>>(dst, ex, mm, ss, E, 1);
  k_edge_agg_scalar<<<cdiv(E, B256), B256, 0, stream>>>(src, dst, ex, ss, bufB, outp, E);
  k_bias_scalar<<<cdiv(N, B256), B256, 0, stream>>>(outp, out_b, outp, N);
}